// OccupancyToTopology_44753559224581
// MI455X (gfx1250) — compile-verified
//
#include <hip/hip_runtime.h>
#include <hip/hip_bf16.h>

typedef __attribute__((ext_vector_type(2))) float v2f;
typedef __attribute__((ext_vector_type(8))) float v8f;

#define OCC_S0 (65 * 65)   // stride of dim0
#define OCC_S1 65          // stride of dim1

// 64^3 voxels, 256 topo values each. One v_wmma_f32_16x16x4_f32 (rank-1 outer
// product hi ⊗ lo) produces all 256 outputs of a voxel.
// 8 waves per block, VPW voxels per wave.
#define VPW 8

__global__ __launch_bounds__(256) void OccupancyToTopology_kernel(
    const float* __restrict__ occ, float* __restrict__ out) {
  const int lane     = threadIdx.x & 31;
  const int waveId   = (int)blockIdx.x * 8 + (threadIdx.x >> 5);
  const int j        = lane & 15;          // nibble index this lane evaluates
  const bool lowHalf = lane < 16;

  const long long nBase = (long long)waveId * VPW;

  #pragma unroll 1
  for (int v = 0; v < VPW; ++v) {
    const long long n = nBase + v;
    // n = i*64*64 + jy*64 + k  (C-order over the 64^3 voxel grid)
    const int i  = (int)(n >> 12);
    const int jy = ((int)n >> 6) & 63;
    const int k  = (int)n & 63;
    const float* base = occ + (long long)i * OCC_S0 + jy * OCC_S1 + k;

    // CORNER_OFFSETS (dx,dy,dz):
    // c0(0,0,0) c1(1,0,0) c2(1,1,0) c3(0,1,0) c4(0,0,1) c5(1,0,1) c6(1,1,1) c7(0,1,1)
    const float p0 = base[0];
    const float p1 = base[OCC_S0];
    const float p2 = base[OCC_S0 + OCC_S1];
    const float p3 = base[OCC_S1];
    const float p4 = base[1];
    const float p5 = base[OCC_S0 + 1];
    const float p6 = base[OCC_S0 + OCC_S1 + 1];
    const float p7 = base[OCC_S1 + 1];

    // lo[j] over corners 0..3 (low nibble of t), hi[j] over corners 4..7.
    const float lo = ((j & 1) ? p0 : 1.0f - p0)
                   * ((j & 2) ? p1 : 1.0f - p1)
                   * ((j & 4) ? p2 : 1.0f - p2)
                   * ((j & 8) ? p3 : 1.0f - p3);
    const float hi = ((j & 1) ? p4 : 1.0f - p4)
                   * ((j & 2) ? p5 : 1.0f - p5)
                   * ((j & 4) ? p6 : 1.0f - p6)
                   * ((j & 8) ? p7 : 1.0f - p7);

    // A (16x4): column K=0 = hi[M] (lanes 0-15 of VGPR0), all other K zero.
    // B (4x16): row K=0 = lo[N] (lanes 0-15 of VGPR0), all other rows zero.
    v2f A, B;
    A.x = lowHalf ? hi : 0.0f;
    A.y = 0.0f;
    B.x = lowHalf ? lo : 0.0f;
    B.y = 0.0f;
    v8f C = {};
    // D[M,N] = hi[M] * lo[N] = topo[t = M*16 + N]
    v8f D = __builtin_amdgcn_wmma_f32_16x16x4_f32(
        /*neg_a=*/false, A, /*neg_b=*/false, B,
        /*c_mod=*/(short)0, C, /*reuse_a=*/false, /*reuse_b=*/false);

    // C/D layout: VGPR r, lanes 0-15 -> M=r, lanes 16-31 -> M=r+8; N = lane&15.
    // Each per-VGPR store is two contiguous 64B runs across the wave.
    float* o = out + (long long)n * 256;
    const int mOff = lowHalf ? 0 : 8;
    #pragma unroll
    for (int r = 0; r < 8; ++r) {
      const int t = (r + mOff) * 16 + j;
      __builtin_nontemporal_store(D[r], o + t);   // 256MB output > 192MB L2: stream it
    }
  }
}

extern "C" void kernel_launch(void* const* d_in, const int* in_sizes, int n_in,
                              void* d_out, int out_size, void* d_ws, size_t ws_size,
                              hipStream_t stream) {
  const float* occ = (const float*)d_in[0];
  float* out = (float*)d_out;
  // 262144 voxels / (8 waves/block * VPW voxels/wave) = 4096 blocks
  OccupancyToTopology_kernel<<<4096, 256, 0, stream>>>(occ, out);
}